// GCN_77988016161258
// MI455X (gfx1250) — compile-verified
//
#include <hip/hip_runtime.h>
#include <math.h>

typedef float v2f __attribute__((ext_vector_type(2)));
typedef float v8f __attribute__((ext_vector_type(8)));

// -------------------- degree / normalization --------------------
__global__ void k_deg_init(float* deg, int n) {
    int i = blockIdx.x * blockDim.x + threadIdx.x;
    if (i < n) deg[i] = 1.0f;                     // self-loop contributes 1
}

__global__ void k_deg_edges(const int* __restrict__ dst, float* deg, int e) {
    int i = blockIdx.x * blockDim.x + threadIdx.x;
    if (i < e) atomicAdd(&deg[dst[i]], 1.0f);
}

__global__ void k_deg_fin(float* deg, int n) {
    int i = blockIdx.x * blockDim.x + threadIdx.x;
    if (i < n) deg[i] = 1.0f / sqrtf(deg[i]);     // deg >= 1 always
}

// -------------------- fp32 WMMA GEMM: C[N,128] = A[N,128] @ B[128,128] ----
// Block = 256 threads = 8 waves; wave w owns column tile n0 = 16*w and keeps
// its whole 128x16 B slab in 64 VGPRs (loaded once per block). The block
// iterates over 8 row tiles; each 16x128 A strip is staged in LDS (row
// stride 132 floats -> bank-conflict-free ds_load_b64 fragment reads).
// Inner loop: 1 ds_load_b64 + 1 v_wmma_f32_16x16x4_f32 per k-step.
#define TILES_PER_BLOCK 8
__global__ __launch_bounds__(256) void k_gemm_wmma(const float* __restrict__ A,
                                                   const float* __restrict__ B,
                                                   float* __restrict__ C_,
                                                   int ntiles) {
    __shared__ float As[16 * 132];
    const int t    = threadIdx.x;
    const int wave = t >> 5;
    const int lane = t & 31;
    const int half = lane >> 4;
    const int r    = lane & 15;
    const int col  = (wave << 4) + r;          // global output column

    // Preload this wave's B slab: breg[i] = { B[4i+2h][col], B[4i+2h+1][col] }
    v2f breg[32];
#pragma unroll
    for (int i = 0; i < 32; ++i) {
        breg[i].x = B[(size_t)(4 * i + 2 * half)     * 128 + col];
        breg[i].y = B[(size_t)(4 * i + 2 * half + 1) * 128 + col];
    }

    for (int it = 0; it < TILES_PER_BLOCK; ++it) {
        int tile = blockIdx.x * TILES_PER_BLOCK + it;   // uniform across block
        if (tile >= ntiles) break;
        const int m0 = tile << 4;

        __syncthreads();   // previous iteration's readers done
#pragma unroll
        for (int v = 0; v < 2; ++v) {          // coalesced 16x128 stage
            int f = t * 8 + v * 4;
            int row = f >> 7, ck = f & 127;
            float4 val = *(const float4*)(A + (size_t)(m0 + row) * 128 + ck);
            *(float4*)(&As[row * 132 + ck]) = val;
        }
        __syncthreads();

        v8f acc = {};
#pragma unroll
        for (int k = 0; k < 128; k += 4) {
            v2f a = *(const v2f*)(&As[r * 132 + k + 2 * half]);
            acc = __builtin_amdgcn_wmma_f32_16x16x4_f32(
                false, a, false, breg[k >> 2], (short)0, acc, false, false);
        }

        float* crow = C_ + (size_t)(m0 + 8 * half) * 128 + col;
#pragma unroll
        for (int j = 0; j < 8; ++j) crow[(size_t)j * 128] = acc[j];
    }
}

// Scalar tail for rows not covered by 16-row tiles (unused when N % 16 == 0).
__global__ void k_gemm_tail(const float* __restrict__ A,
                            const float* __restrict__ B,
                            float* __restrict__ C_, int mstart, int mcount) {
    int tid = blockIdx.x * blockDim.x + threadIdx.x;
    if (tid >= mcount * 128) return;
    int m = mstart + tid / 128, n = tid & 127;
    float s = 0.f;
    for (int k = 0; k < 128; ++k) s += A[(size_t)m * 128 + k] * B[(size_t)k * 128 + n];
    C_[(size_t)m * 128 + n] = s;
}

// -------------------- aggregation --------------------
// Self-loop term also initializes the accumulation buffer (plain store).
__global__ void k_self_init(const float* __restrict__ hlin,
                            const float* __restrict__ dinv,
                            float* __restrict__ out, int n) {
    int tid = blockIdx.x * blockDim.x + threadIdx.x;
    int ni = tid >> 5;
    if (ni >= n) return;
    int f = (tid & 31) << 2;
    float w = dinv[ni]; w *= w;
    float4 v = *(const float4*)(hlin + (size_t)ni * 128 + f);
    float4 o = { v.x * w, v.y * w, v.z * w, v.w * w };
    *(float4*)(out + (size_t)ni * 128 + f) = o;
}

// One wave per edge; lane handles 4 features -> 4 f32 L2 atomics.
__global__ void k_edge_agg(const int* __restrict__ src, const int* __restrict__ dst,
                           const float* __restrict__ dinv,
                           const float* __restrict__ hlin,
                           float* __restrict__ out, int e) {
    int tid = blockIdx.x * blockDim.x + threadIdx.x;
    int ei = tid >> 5;
    if (ei >= e) return;
    int f = (tid & 31) << 2;
    int s = src[ei], d = dst[ei];
    float nw = dinv[s] * dinv[d];
    float4 v = *(const float4*)(hlin + (size_t)s * 128 + f);
    float* o = out + (size_t)d * 128 + f;
    atomicAdd(o + 0, v.x * nw);
    atomicAdd(o + 1, v.y * nw);
    atomicAdd(o + 2, v.z * nw);
    atomicAdd(o + 3, v.w * nw);
}

__global__ void k_bias_relu(float* __restrict__ h, const float* __restrict__ b, int total) {
    int tid = blockIdx.x * blockDim.x + threadIdx.x;
    if (tid >= total) return;
    float v = h[tid] + b[tid & 127];
    h[tid] = v > 0.f ? v : 0.f;
}

// -------------------- classifier (accumulated per layer) --------------------
__global__ void k_zinit(float* __restrict__ z, const float* __restrict__ bout,
                        int n, int C) {
    int tid = blockIdx.x * blockDim.x + threadIdx.x;
    if (tid < n * C) z[tid] = bout[tid % C];
}

__global__ void k_zacc(const float* __restrict__ h, const float* __restrict__ Wout,
                       int rowoff, float* __restrict__ z, int n, int C) {
    __shared__ float w[128 * 40];
    const int nelem = 128 * C;
    for (int i = threadIdx.x; i < nelem; i += blockDim.x)
        w[i] = Wout[(size_t)rowoff * C + i];
    __syncthreads();
    int tid = blockIdx.x * blockDim.x + threadIdx.x;
    if (tid >= n * C) return;
    int node = tid / C, c = tid % C;
    const float* hr = h + (size_t)node * 128;
    float s = 0.f;
#pragma unroll 8
    for (int k = 0; k < 128; ++k) s += hr[k] * w[k * C + c];
    z[tid] += s;
}

__global__ void k_logsoftmax(const float* __restrict__ z, float* __restrict__ out,
                             int n, int C) {
    int i = blockIdx.x * blockDim.x + threadIdx.x;
    if (i >= n) return;
    const float* zr = z + (size_t)i * C;
    float m = -1e30f;
    for (int c = 0; c < C; ++c) m = fmaxf(m, zr[c]);
    float s = 0.f;
    for (int c = 0; c < C; ++c) s += expf(zr[c] - m);
    float l = m + logf(s);
    float* o = out + (size_t)i * C;
    for (int c = 0; c < C; ++c) o[c] = zr[c] - l;
}

// -------------------- host orchestration --------------------
extern "C" void kernel_launch(void* const* d_in, const int* in_sizes, int n_in,
                              void* d_out, int out_size, void* d_ws, size_t ws_size,
                              hipStream_t stream) {
    const float* x    = (const float*)d_in[0];
    const int*   ei   = (const int*)d_in[1];
    const float* W[3] = {(const float*)d_in[2], (const float*)d_in[4], (const float*)d_in[6]};
    const float* b[3] = {(const float*)d_in[3], (const float*)d_in[5], (const float*)d_in[7]};
    const float* Wout = (const float*)d_in[8];
    const float* bout = (const float*)d_in[9];

    const int Nn = in_sizes[0] / 128;
    const int Ee = in_sizes[1] / 2;
    const int C  = in_sizes[9];
    const int* src = ei;
    const int* dst = ei + Ee;

    // workspace layout: dinv | lin | bufB | bufC | z   (~170 MB for N=100k)
    char* base = (char*)d_ws;
    size_t off = ((size_t)Nn * 4 + 255) & ~(size_t)255;
    size_t SZ  = (size_t)Nn * 128 * 4;
    float* dinv = (float*)base;
    float* lin  = (float*)(base + off);
    float* bufB = (float*)(base + off + SZ);
    float* bufC = (float*)(base + off + 2 * SZ);
    float* z    = (float*)(base + off + 3 * SZ);

    const int TPB = 256;
    // normalization coefficients
    k_deg_init<<<(Nn + TPB - 1) / TPB, TPB, 0, stream>>>(dinv, Nn);
    k_deg_edges<<<(Ee + TPB - 1) / TPB, TPB, 0, stream>>>(dst, dinv, Ee);
    k_deg_fin<<<(Nn + TPB - 1) / TPB, TPB, 0, stream>>>(dinv, Nn);
    // classifier accumulator starts at bias
    k_zinit<<<((size_t)Nn * C + TPB - 1) / TPB, TPB, 0, stream>>>(z, bout, Nn, C);

    const float* prev = x;
    float* curs[3] = {bufB, bufC, bufB};
    const int mt = Nn / 16, rem = Nn % 16;
    const int gblocks = (mt + TILES_PER_BLOCK - 1) / TILES_PER_BLOCK;

    for (int li = 0; li < 3; ++li) {
        float* cur = curs[li];
        k_gemm_wmma<<<gblocks, 256, 0, stream>>>(prev, W[li], lin, mt);
        if (rem)
            k_gemm_tail<<<(rem * 128 + TPB - 1) / TPB, TPB, 0, stream>>>(
                prev, W[li], lin, mt * 16, rem);
        k_self_init<<<((size_t)Nn * 32 + TPB - 1) / TPB, TPB, 0, stream>>>(lin, dinv, cur, Nn);
        k_edge_agg<<<((size_t)Ee * 32 + TPB - 1) / TPB, TPB, 0, stream>>>(src, dst, dinv, lin, cur, Ee);
        k_bias_relu<<<((size_t)Nn * 128 + TPB - 1) / TPB, TPB, 0, stream>>>(cur, b[li], Nn * 128);
        k_zacc<<<((size_t)Nn * C + TPB - 1) / TPB, TPB, 0, stream>>>(cur, Wout, 128 * li, z, Nn, C);
        prev = cur;
    }

    k_logsoftmax<<<(Nn + TPB - 1) / TPB, TPB, 0, stream>>>(z, (float*)d_out, Nn, C);
}